// PaiNNInteraction_37220186587475
// MI455X (gfx1250) — compile-verified
//
#include <hip/hip_runtime.h>

typedef __attribute__((ext_vector_type(2))) float v2f;
typedef __attribute__((ext_vector_type(8))) float v8f;

#define PDIM 128

// ---------------------------------------------------------------------------
// Init: s_out = s (agg_s accumulates into it), agg_v workspace = 0.
// Must run every launch (harness does not re-poison between replays).
// ---------------------------------------------------------------------------
__global__ __launch_bounds__(256) void painn_init(const float* __restrict__ s,
                                                  float* __restrict__ s_out,
                                                  float* __restrict__ aggv,
                                                  int n_s4, int n_a4) {
  int i = blockIdx.x * blockDim.x + threadIdx.x;
  int stride = gridDim.x * blockDim.x;
  for (int j = i; j < n_s4; j += stride)
    ((float4*)s_out)[j] = ((const float4*)s)[j];
  float4 z = make_float4(0.f, 0.f, 0.f, 0.f);
  for (int j = i; j < n_a4; j += stride)
    ((float4*)aggv)[j] = z;
}

// ---------------------------------------------------------------------------
// Fused node GEMM: [q | k | v_s | v_v] = s @ [Wq | Wkv] + [bq | bkv]
// One wave per 16x16 output tile, f32 WMMA, K=128 in 32 steps of K=4.
// 512 output columns: 0-127 -> q, 128-255 -> k, 256-383 -> v_s, 384-511 -> v_v
// ---------------------------------------------------------------------------
__global__ __launch_bounds__(128) void painn_node_gemm(
    const float* __restrict__ s,
    const float* __restrict__ Wq,  const float* __restrict__ bq,
    const float* __restrict__ Wkv, const float* __restrict__ bkv,
    float* __restrict__ q, float* __restrict__ k,
    float* __restrict__ vs, float* __restrict__ vv,
    int nTiles) {
  const int nColTiles = 512 / 16;  // 32
  int wave = threadIdx.x >> 5;
  int lane = threadIdx.x & 31;
  int tile = blockIdx.x * 4 + wave;
  if (tile >= nTiles) return;  // wave-uniform: EXEC stays all-1s for WMMA
  int rowTile = tile / nColTiles;
  int colTile = tile % nColTiles;
  int rowBase = rowTile * 16;
  int cGlobal = colTile * 16;
  int half = lane >> 4;       // 0: lanes 0-15, 1: lanes 16-31
  int l16  = lane & 15;
  int col  = cGlobal + l16;   // 0..511

  // Weight/bias segment (uniform per wave: tiles are 16-aligned)
  const float* W; const float* b; int wcol; int ldw;
  if (cGlobal < PDIM) { W = Wq;  b = bq;  wcol = col;        ldw = PDIM;     }
  else                { W = Wkv; b = bkv; wcol = col - PDIM; ldw = 3 * PDIM; }

  v8f acc = {};
  const float* srow = s + (size_t)(rowBase + l16) * PDIM;
  for (int k0 = 0; k0 < PDIM; k0 += 4) {
    // A 16x4: lane row = l16; VGPR0/1 = K pair {k0+2*half, k0+2*half+1}
    float2 av = *(const float2*)(srow + k0 + 2 * half);
    v2f a; a.x = av.x; a.y = av.y;
    // B 4x16: lane col = l16; VGPR0/1 = rows {k0+2*half, k0+2*half+1}
    v2f bf;
    bf.x = W[(size_t)(k0 + 2 * half)     * ldw + wcol];
    bf.y = W[(size_t)(k0 + 2 * half + 1) * ldw + wcol];
    acc = __builtin_amdgcn_wmma_f32_16x16x4_f32(false, a, false, bf,
                                                (short)0, acc, false, false);
  }
  float bias = b[wcol];

  float* out;
  if      (cGlobal < 128) out = q;
  else if (cGlobal < 256) out = k;
  else if (cGlobal < 384) out = vs;
  else                    out = vv;
  int oc = col & (PDIM - 1);
  for (int r = 0; r < 8; ++r) {
    int row = rowBase + r + 8 * half;  // C/D layout: M = r + 8*(lane>=16)
    out[(size_t)row * PDIM + oc] = acc[r] + bias;
  }
}

// ---------------------------------------------------------------------------
// Edge kernel: one wave per edge. Gathers hit L2 (node data ~175MB < 192MB L2).
// w = dot(q[row], k[col]) via wave32 shuffle reduction; scatter with f32 atomics.
// ---------------------------------------------------------------------------
__global__ __launch_bounds__(256) void painn_edge(
    const float* __restrict__ q, const float* __restrict__ k,
    const float* __restrict__ vs, const float* __restrict__ vv,
    const float* __restrict__ v,
    const int* __restrict__ rowIdx, const int* __restrict__ colIdx,
    float* __restrict__ s_out, float* __restrict__ aggv, int E) {
  int e = blockIdx.x * 8 + (threadIdx.x >> 5);
  if (e >= E) return;
  int lane = threadIdx.x & 31;
  int r = rowIdx[e];
  int c = colIdx[e];
  int d = lane * 4;  // 4 dims of 128 per lane

  float4 qv = *(const float4*)(q + (size_t)r * PDIM + d);
  float4 kv = *(const float4*)(k + (size_t)c * PDIM + d);
  float p = qv.x * kv.x + qv.y * kv.y + qv.z * kv.z + qv.w * kv.w;
  p += __shfl_xor(p, 16, 32);
  p += __shfl_xor(p, 8, 32);
  p += __shfl_xor(p, 4, 32);
  p += __shfl_xor(p, 2, 32);
  p += __shfl_xor(p, 1, 32);
  float w = p;

  float4 vsv = *(const float4*)(vs + (size_t)c * PDIM + d);
  float* so = s_out + (size_t)r * PDIM + d;
  atomicAdd(so + 0, w * vsv.x);
  atomicAdd(so + 1, w * vsv.y);
  atomicAdd(so + 2, w * vsv.z);
  atomicAdd(so + 3, w * vsv.w);

  float4 vvv = *(const float4*)(vv + (size_t)c * PDIM + d);
  const float* vc = v + (size_t)c * 3 * PDIM;
  float* av = aggv + (size_t)r * 3 * PDIM;
#pragma unroll
  for (int comp = 0; comp < 3; ++comp) {
    float4 vval = *(const float4*)(vc + comp * PDIM + d);
    float* ap = av + comp * PDIM + d;
    atomicAdd(ap + 0, w * vvv.x * vval.x);
    atomicAdd(ap + 1, w * vvv.y * vval.y);
    atomicAdd(ap + 2, w * vvv.z * vval.z);
    atomicAdd(ap + 3, w * vvv.w * vval.w);
  }
}

// ---------------------------------------------------------------------------
// v_out = v + agg_v @ Wvec + bvec, treated as GEMM M=3N, K=128, N=128.
// ---------------------------------------------------------------------------
__global__ __launch_bounds__(128) void painn_vec_gemm(
    const float* __restrict__ aggv, const float* __restrict__ Wvec,
    const float* __restrict__ bvec, const float* __restrict__ v,
    float* __restrict__ vout, int nTiles) {
  const int nColTiles = PDIM / 16;  // 8
  int wave = threadIdx.x >> 5;
  int lane = threadIdx.x & 31;
  int tile = blockIdx.x * 4 + wave;
  if (tile >= nTiles) return;  // wave-uniform
  int rowTile = tile / nColTiles;
  int colTile = tile % nColTiles;
  int rowBase = rowTile * 16;
  int half = lane >> 4;
  int l16  = lane & 15;
  int col  = colTile * 16 + l16;

  v8f acc = {};
  const float* arow = aggv + (size_t)(rowBase + l16) * PDIM;
  for (int k0 = 0; k0 < PDIM; k0 += 4) {
    float2 av2 = *(const float2*)(arow + k0 + 2 * half);
    v2f a; a.x = av2.x; a.y = av2.y;
    v2f bf;
    bf.x = Wvec[(size_t)(k0 + 2 * half)     * PDIM + col];
    bf.y = Wvec[(size_t)(k0 + 2 * half + 1) * PDIM + col];
    acc = __builtin_amdgcn_wmma_f32_16x16x4_f32(false, a, false, bf,
                                                (short)0, acc, false, false);
  }
  float bias = bvec[col];
  for (int r = 0; r < 8; ++r) {
    size_t row = (size_t)rowBase + r + 8 * half;
    size_t idx = row * PDIM + col;
    vout[idx] = v[idx] + acc[r] + bias;
  }
}

// ---------------------------------------------------------------------------
extern "C" void kernel_launch(void* const* d_in, const int* in_sizes, int n_in,
                              void* d_out, int out_size, void* d_ws, size_t ws_size,
                              hipStream_t stream) {
  const float* s    = (const float*)d_in[0];
  const float* v    = (const float*)d_in[1];
  const int*   ei   = (const int*)d_in[2];
  const float* Wq   = (const float*)d_in[3];
  const float* bq   = (const float*)d_in[4];
  const float* Wkv  = (const float*)d_in[5];
  const float* bkv  = (const float*)d_in[6];
  const float* Wvec = (const float*)d_in[7];
  const float* bvec = (const float*)d_in[8];

  int N = in_sizes[0] / PDIM;  // 50000
  int E = in_sizes[2] / 2;     // 600000

  float* s_out = (float*)d_out;                 // [N,128]
  float* v_out = s_out + (size_t)N * PDIM;      // [N,3,128]

  // Workspace: q, k, v_s, v_v [N,128] each, then agg_v [N,3,128]
  float* q    = (float*)d_ws;
  float* k    = q  + (size_t)N * PDIM;
  float* vs   = k  + (size_t)N * PDIM;
  float* vv   = vs + (size_t)N * PDIM;
  float* aggv = vv + (size_t)N * PDIM;

  const int* rowIdx = ei;       // edge_index[0]
  const int* colIdx = ei + E;   // edge_index[1]

  painn_init<<<2048, 256, 0, stream>>>(s, s_out, aggv,
                                       N * PDIM / 4, N * 3 * PDIM / 4);

  int tilesA = (N / 16) * (512 / 16);           // 100000 tiles
  painn_node_gemm<<<(tilesA + 3) / 4, 128, 0, stream>>>(
      s, Wq, bq, Wkv, bkv, q, k, vs, vv, tilesA);

  painn_edge<<<(E + 7) / 8, 256, 0, stream>>>(
      q, k, vs, vv, v, rowIdx, colIdx, s_out, aggv, E);

  int tilesC = (N * 3 / 16) * (PDIM / 16);      // 75000 tiles
  painn_vec_gemm<<<(tilesC + 3) / 4, 128, 0, stream>>>(
      aggv, Wvec, bvec, v, v_out, tilesC);
}